// GATConv_1065151889892
// MI455X (gfx1250) — compile-verified
//
#include <hip/hip_runtime.h>
#include <hip/hip_bf16.h>
#include <math.h>

#define N_NODES 50000
#define N_EDGES 800000
#define IN_CH   128
#define OUT_CH  64
#define HEADS   8

typedef __attribute__((ext_vector_type(2))) float v2f;
typedef __attribute__((ext_vector_type(8))) float v8f;

// ---------------------------------------------------------------------------
// Zero fill
// ---------------------------------------------------------------------------
__global__ __launch_bounds__(256) void zero_f32(float* __restrict__ p, int n) {
    int i = blockIdx.x * 256 + threadIdx.x;
    if (i < n) p[i] = 0.0f;
}

// ---------------------------------------------------------------------------
// h = x @ W  with V_WMMA_F32_16X16X4_F32 (fp32 in/acc — reference precision).
// One wave per 16-row tile, 4 WMMA column tiles (all 64 outputs) sharing the
// A fragment.
//
// W is staged in LDS in a K-PAIR-INTERLEAVED layout:
//     Wlds[(k>>1)*WSTRIDE + 2*col + (k&1)] = W[k][col]
// so a B fragment {W[kk][col], W[kk+1][col]} is ONE aligned ds_load_b64 into
// an even VGPR pair — no VALU repacking before the WMMA. WSTRIDE=160 floats
// (640 B ≡ 32 banks mod 64) makes lanes 0-15 (row kk = k) and lanes 16-31
// (row kk = k+2) of a single ds_load_b64 hit disjoint bank halves.
//
// fp32 WMMA VGPR layouts (ISA 7.12.2):
//   A (16x4):  lane l / l+16 : v0 = A[l][k+2*half], v1 = A[l][k+2*half+1]
//   B (4x16):  lane = col,     v0 = B[k+2*half][col], v1 = B[k+2*half+1][col]
//   C/D(16x16): vgpr v: lanes 0-15 -> row v, lanes 16-31 -> row v+8; col=l
// ---------------------------------------------------------------------------
#define WSTRIDE 160   // floats per interleaved K-pair row (64*2 data + 32 skew)

__global__ __launch_bounds__(256) void gemm_wmma_f32(
        const float* __restrict__ x, const float* __restrict__ W,
        float* __restrict__ h) {
    __shared__ float Wlds[(IN_CH / 2) * WSTRIDE];  // 40 KB of 320 KB/WGP
    for (int i = threadIdx.x; i < IN_CH * OUT_CH; i += 256) {
        const int k = i >> 6;        // 0..127
        const int col = i & 63;      // 0..63
        Wlds[(k >> 1) * WSTRIDE + 2 * col + (k & 1)] = W[i];
    }
    __syncthreads();

    const int wave = blockIdx.x * 8 + (threadIdx.x >> 5);
    const int lane = threadIdx.x & 31;
    const int numRowTiles = N_NODES / 16;  // 3125, exact
    if (wave >= numRowTiles) return;       // wave-uniform: EXEC all-1s inside

    const int row0 = wave * 16;
    const int half = lane >> 4;   // which K-pair of the k-step this lane holds
    const int l    = lane & 15;   // row (A) / col (B,C,D) index

    const float* xrow = x + (size_t)(row0 + l) * IN_CH;

    v8f c0 = {}, c1 = {}, c2 = {}, c3 = {};
    #pragma unroll 4
    for (int k = 0; k < IN_CH; k += 4) {
        const int kk = k + 2 * half;                       // even
        v2f a = *(const v2f*)(xrow + kk);                  // global_load_b64
        const float* wb = &Wlds[(kk >> 1) * WSTRIDE + 2 * l];
        v2f b0 = *(const v2f*)(wb +  0);                   // ds_load_b64 each,
        v2f b1 = *(const v2f*)(wb + 32);                   // bank-conflict-free
        v2f b2 = *(const v2f*)(wb + 64);
        v2f b3 = *(const v2f*)(wb + 96);
        c0 = __builtin_amdgcn_wmma_f32_16x16x4_f32(false, a, false, b0, (short)0, c0, false, false);
        c1 = __builtin_amdgcn_wmma_f32_16x16x4_f32(false, a, false, b1, (short)0, c1, false, false);
        c2 = __builtin_amdgcn_wmma_f32_16x16x4_f32(false, a, false, b2, (short)0, c2, false, false);
        c3 = __builtin_amdgcn_wmma_f32_16x16x4_f32(false, a, false, b3, (short)0, c3, false, false);
    }

    float* hb = h + (size_t)row0 * OUT_CH;
    #pragma unroll
    for (int v = 0; v < 8; ++v) {
        const int r = (v + 8 * half) * OUT_CH + l;
        hb[r +  0] = c0[v];
        hb[r + 16] = c1[v];
        hb[r + 32] = c2[v];
        hb[r + 48] = c3[v];
    }
}

// ---------------------------------------------------------------------------
// Per-edge logit = leaky_relu(h[ei].a1 + h[ej].a2, 0.2); block-max partials.
// ---------------------------------------------------------------------------
__global__ __launch_bounds__(256) void edge_logits(
        const float* __restrict__ h, const int* __restrict__ eidx,
        const float* __restrict__ attn, float* __restrict__ logits,
        float* __restrict__ partialMax) {
    __shared__ float a1[OUT_CH], a2[OUT_CH];
    __shared__ float red[256];
    if (threadIdx.x < OUT_CH) {
        a1[threadIdx.x] = attn[threadIdx.x];
        a2[threadIdx.x] = attn[OUT_CH + threadIdx.x];
    }
    __syncthreads();

    const int e = blockIdx.x * 256 + threadIdx.x;
    float v = -INFINITY;
    if (e < N_EDGES) {
        const int di = eidx[e];
        const int sj = eidx[N_EDGES + e];
        const float4* hi = (const float4*)(h + (size_t)di * OUT_CH);
        const float4* hj = (const float4*)(h + (size_t)sj * OUT_CH);
        float s = 0.0f;
        #pragma unroll
        for (int q = 0; q < OUT_CH / 4; ++q) {
            float4 u = hi[q];
            float4 w = hj[q];
            s += u.x * a1[4*q+0] + u.y * a1[4*q+1] + u.z * a1[4*q+2] + u.w * a1[4*q+3];
            s += w.x * a2[4*q+0] + w.y * a2[4*q+1] + w.z * a2[4*q+2] + w.w * a2[4*q+3];
        }
        const float logit = (s > 0.0f) ? s : 0.2f * s;
        logits[e] = logit;
        v = logit;
    }
    red[threadIdx.x] = v;
    __syncthreads();
    for (int st = 128; st > 0; st >>= 1) {
        if (threadIdx.x < st)
            red[threadIdx.x] = fmaxf(red[threadIdx.x], red[threadIdx.x + st]);
        __syncthreads();
    }
    if (threadIdx.x == 0) partialMax[blockIdx.x] = red[0];
}

// ---------------------------------------------------------------------------
// Single-block reductions over the 3125 block partials.
// ---------------------------------------------------------------------------
__global__ __launch_bounds__(1024) void reduce_max_k(
        const float* __restrict__ partial, int n, float* __restrict__ gmax) {
    __shared__ float s[1024];
    float v = -INFINITY;
    for (int i = threadIdx.x; i < n; i += 1024) v = fmaxf(v, partial[i]);
    s[threadIdx.x] = v;
    __syncthreads();
    for (int st = 512; st > 0; st >>= 1) {
        if (threadIdx.x < st) s[threadIdx.x] = fmaxf(s[threadIdx.x], s[threadIdx.x + st]);
        __syncthreads();
    }
    if (threadIdx.x == 0) gmax[0] = s[0];
}

__global__ __launch_bounds__(256) void edge_exp(
        float* __restrict__ logits, const float* __restrict__ gmax,
        float* __restrict__ partialSum) {
    __shared__ float red[256];
    const int e = blockIdx.x * 256 + threadIdx.x;
    float ex = 0.0f;
    if (e < N_EDGES) {
        ex = expf(logits[e] - gmax[0]);
        logits[e] = ex;   // overwrite in place with exp value
    }
    red[threadIdx.x] = ex;
    __syncthreads();
    for (int st = 128; st > 0; st >>= 1) {
        if (threadIdx.x < st) red[threadIdx.x] += red[threadIdx.x + st];
        __syncthreads();
    }
    if (threadIdx.x == 0) partialSum[blockIdx.x] = red[0];
}

__global__ __launch_bounds__(1024) void reduce_sum_k(
        const float* __restrict__ partial, int n, float* __restrict__ invsum) {
    __shared__ float s[1024];
    float v = 0.0f;
    for (int i = threadIdx.x; i < n; i += 1024) v += partial[i];
    s[threadIdx.x] = v;
    __syncthreads();
    for (int st = 512; st > 0; st >>= 1) {
        if (threadIdx.x < st) s[threadIdx.x] += s[threadIdx.x + st];
        __syncthreads();
    }
    if (threadIdx.x == 0) invsum[0] = 1.0f / s[0];
}

// ---------------------------------------------------------------------------
// Scatter: h_head[ei] += h[ej] * alpha. 64 threads/edge -> coalesced channels;
// atomics land in the L2-resident 12.8MB accumulator.
// ---------------------------------------------------------------------------
__global__ __launch_bounds__(256) void scatter_add(
        const float* __restrict__ h, const int* __restrict__ eidx,
        const float* __restrict__ expv, const float* __restrict__ invsum,
        float* __restrict__ hhead) {
    const long long gid = (long long)blockIdx.x * 256 + threadIdx.x;
    const int e = (int)(gid >> 6);
    const int c = (int)(gid & 63);
    if (e >= N_EDGES) return;
    const int di = eidx[e];
    const int sj = eidx[N_EDGES + e];
    const float alpha = expv[e] * invsum[0];
    const float val = h[(size_t)sj * OUT_CH + c] * alpha;
    atomicAdd(&hhead[(size_t)di * OUT_CH + c], val);
}

// ---------------------------------------------------------------------------
// out[n, t*64+c] = hhead[n, c] for t=0..7 : 1 read, 8 coalesced writes.
// ---------------------------------------------------------------------------
__global__ __launch_bounds__(256) void tile_out(
        const float* __restrict__ hhead, float* __restrict__ out) {
    const int i = blockIdx.x * 256 + threadIdx.x;   // over N_NODES*64 = 3.2M
    if (i >= N_NODES * OUT_CH) return;
    const int n = i >> 6;
    const int c = i & 63;
    const float v = hhead[i];
    float* o = out + (size_t)n * (OUT_CH * HEADS) + c;
    #pragma unroll
    for (int t = 0; t < HEADS; ++t) o[t * OUT_CH] = v;
}

// ---------------------------------------------------------------------------
extern "C" void kernel_launch(void* const* d_in, const int* in_sizes, int n_in,
                              void* d_out, int out_size, void* d_ws, size_t ws_size,
                              hipStream_t stream) {
    const float* x    = (const float*)d_in[0];  // [50000,128]
    const int*   eidx = (const int*)  d_in[1];  // [2,800000]
    const float* W    = (const float*)d_in[2];  // [128,64]
    const float* attn = (const float*)d_in[3];  // [128,1]
    float* out = (float*)d_out;                 // [50000,512]

    // workspace layout (floats)
    float* ws      = (float*)d_ws;
    float* h       = ws;                         // 3,200,000
    float* hhead   = ws + 3200000;               // 3,200,000
    float* logits  = ws + 6400000;               //   800,000 (logit -> exp)
    float* partial = ws + 7200000;               //     3,125
    float* gmax    = ws + 7203328;               //         1
    float* invsum  = ws + 7203329;               //         1

    const int edgeBlocks = N_EDGES / 256;        // 3125 (exact)

    // 1. zero accumulator
    zero_f32<<<(N_NODES * OUT_CH + 255) / 256, 256, 0, stream>>>(hhead, N_NODES * OUT_CH);
    // 2. h = x @ W  (fp32 WMMA)
    gemm_wmma_f32<<<(N_NODES / 16 + 7) / 8, 256, 0, stream>>>(x, W, h);
    // 3. per-edge logits + block max
    edge_logits<<<edgeBlocks, 256, 0, stream>>>(h, eidx, attn, logits, partial);
    // 4. global max
    reduce_max_k<<<1, 1024, 0, stream>>>(partial, edgeBlocks, gmax);
    // 5. exp(logit - max) + block sums
    edge_exp<<<edgeBlocks, 256, 0, stream>>>(logits, gmax, partial);
    // 6. 1/sum
    reduce_sum_k<<<1, 1024, 0, stream>>>(partial, edgeBlocks, invsum);
    // 7. scatter-add messages
    scatter_add<<<(int)(((long long)N_EDGES * OUT_CH) / 256), 256, 0, stream>>>(
        h, eidx, logits, invsum, hhead);
    // 8. tile heads into output
    tile_out<<<(N_NODES * OUT_CH + 255) / 256, 256, 0, stream>>>(hhead, out);
}